// EGNNLayer_4724464025870
// MI455X (gfx1250) — compile-verified
//
#include <hip/hip_runtime.h>

// EGNN layer for MI455X (gfx1250, wave32, WMMA).
// B=2, N=768, D=32, DM=64, K=16. All math in f32 via V_WMMA_F32_16X16X4_F32.

#define B_   2
#define N_   768
#define D_   32
#define DM_  64
#define K_   16
#define JT_  (N_ / 16)      // 48 j-tiles per node
#define STRIDE_ 66          // padded LDS row stride (even -> b64 aligned, no bank conflicts)

typedef __attribute__((ext_vector_type(2))) float v2f;
typedef __attribute__((ext_vector_type(8))) float v8f;

// D(16x16,f32) = A(16x4,f32) * B(4x16,f32) + C
__device__ __forceinline__ v8f wmma4(v2f a, v2f b, v8f c) {
  return __builtin_amdgcn_wmma_f32_16x16x4_f32(false, a, false, b, (short)0, c,
                                               false, false);
}

// Wave-local LDS ordering: DS ops from one wave complete in order; this wait
// guarantees prior ds_stores are visible to subsequent ds_loads of the wave.
__device__ __forceinline__ void lds_fence_wave() {
  asm volatile("s_wait_dscnt 0" ::: "memory");
}

// -------------------------------------------------------------------------
// Shared edge-MLP front end: f2 = L_e1(L_e0(edge_feat)) for a 16-edge tile.
// edge_feat row r (r = lane%16) = [ h[b,j_r](32) | h[b,i](32) | d2_r ], K padded 65->68.
// On return: activated e1 output is in out[4] (WMMA C-layout) AND staged in st.
// -------------------------------------------------------------------------
__device__ __forceinline__ void edge_e0e1(
    const float* __restrict__ hj, const float* __restrict__ hi, float d2,
    const float* __restrict__ e0W, const float* __restrict__ e0b,
    const float* __restrict__ e0s, const float* __restrict__ e0t,
    const float* __restrict__ e1W, const float* __restrict__ e1b,
    const float* __restrict__ e1s, const float* __restrict__ e1t,
    float* st, int grp, int ln, v8f out[4])
{
  // ---- GEMM1: (16 x 68pad) @ e0W(65 x 64) ----
  v8f acc[4] = {};
#pragma unroll
  for (int s = 0; s < 17; ++s) {
    const int k0 = 4 * s + grp * 2;           // A layout: lane holds K = k0, k0+1
    const float vj0 = hj[k0 & 31],       vj1 = hj[(k0 + 1) & 31];
    const float vi0 = hi[k0 & 31],       vi1 = hi[(k0 + 1) & 31];
    const float a0 = (k0 < 32) ? vj0
                   : (k0 < 64) ? vi0
                   : (k0 == 64) ? d2 : 0.0f;
    const float a1 = (k0 + 1 < 32) ? vj1
                   : (k0 + 1 < 64) ? vi1
                   : (k0 + 1 == 64) ? d2 : 0.0f;
    v2f A; A[0] = a0; A[1] = a1;
    const int kk0 = (k0     < 64) ? k0     : 64;   // clamp pad rows (A==0 there)
    const int kk1 = (k0 + 1 < 64) ? k0 + 1 : 64;
#pragma unroll
    for (int t = 0; t < 4; ++t) {
      const int c = 16 * t + ln;
      v2f Bf; Bf[0] = e0W[kk0 * DM_ + c]; Bf[1] = e0W[kk1 * DM_ + c];
      acc[t] = wmma4(A, Bf, acc[t]);
    }
  }
  // affine + ReLU, stage (C-layout row M = v + 8*grp, col = 16t+ln)
#pragma unroll
  for (int t = 0; t < 4; ++t) {
    const int c = 16 * t + ln;
    const float bb = e0b[c], sc = e0s[c], sh = e0t[c];
#pragma unroll
    for (int v = 0; v < 8; ++v) {
      const float f = fmaxf((acc[t][v] + bb) * sc + sh, 0.0f);
      st[(v + 8 * grp) * STRIDE_ + c] = f;
    }
  }
  lds_fence_wave();

  // ---- GEMM2: (16 x 64) @ e1W(64 x 64), A re-read from LDS in A-layout ----
  v8f acc2[4] = {};
#pragma unroll
  for (int s = 0; s < 16; ++s) {
    const int k0 = 4 * s + grp * 2;
    v2f A = *(const v2f*)&st[ln * STRIDE_ + k0];
#pragma unroll
    for (int t = 0; t < 4; ++t) {
      const int c = 16 * t + ln;
      v2f Bf; Bf[0] = e1W[k0 * DM_ + c]; Bf[1] = e1W[(k0 + 1) * DM_ + c];
      acc2[t] = wmma4(A, Bf, acc2[t]);
    }
  }
#pragma unroll
  for (int t = 0; t < 4; ++t) {
    const int c = 16 * t + ln;
    const float bb = e1b[c], sc = e1s[c], sh = e1t[c];
#pragma unroll
    for (int v = 0; v < 8; ++v) {
      const float f = fmaxf((acc2[t][v] + bb) * sc + sh, 0.0f);
      out[t][v] = f;
      st[(v + 8 * grp) * STRIDE_ + c] = f;
    }
  }
  lds_fence_wave();
}

// -------------------------------------------------------------------------
// Kernel 1: per-node KNN (K=16 smallest d2, ties -> smaller index) ; wave per node
// -------------------------------------------------------------------------
__global__ __launch_bounds__(128) void egnn_knn(const float* __restrict__ x,
                                                int* __restrict__ idx)
{
  const int wid = threadIdx.x >> 5, lane = threadIdx.x & 31;
  const int g = blockIdx.x * 4 + wid;           // node id in [0, B*N)
  const int b = g / N_, i = g - b * N_;
  const float* xi = x + (size_t)(b * N_ + i) * 3;
  const float xi0 = xi[0], xi1 = xi[1], xi2 = xi[2];

  float dv[24];                                 // 768 = 24 * 32 lanes
#pragma unroll
  for (int q = 0; q < 24; ++q) {
    const int j = q * 32 + lane;
    const float* xj = x + (size_t)(b * N_ + j) * 3;
    const float a = xj[0] - xi0, bb = xj[1] - xi1, cc = xj[2] - xi2;
    dv[q] = a * a + bb * bb + cc * cc;
  }
  for (int r = 0; r < K_; ++r) {
    float best = 3.4e38f; int bj = 0x7fffffff;
#pragma unroll
    for (int q = 0; q < 24; ++q) {
      const int j = q * 32 + lane;
      if (dv[q] < best || (dv[q] == best && j < bj)) { best = dv[q]; bj = j; }
    }
#pragma unroll
    for (int m = 1; m < 32; m <<= 1) {
      const float ov = __shfl_xor(best, m, 32);
      const int   oj = __shfl_xor(bj, m, 32);
      if (ov < best || (ov == best && oj < bj)) { best = ov; bj = oj; }
    }
    if (lane == 0) idx[(size_t)(b * N_ + i) * K_ + r] = bj;
#pragma unroll
    for (int q = 0; q < 24; ++q)
      if (q * 32 + lane == bj) dv[q] = 3.4e38f;
  }
}

// -------------------------------------------------------------------------
// Kernel 2: full NxN edge pass. Wave per (b, i, 16-j tile).
// Computes w = L_c1(L_c0(m_edge)) and writes the tile-partial sum_j w*x_diff.
// -------------------------------------------------------------------------
__global__ __launch_bounds__(256) void egnn_edge_pass(
    const float* __restrict__ x, const float* __restrict__ h,
    const float* __restrict__ e0W, const float* __restrict__ e0b,
    const float* __restrict__ e0s, const float* __restrict__ e0t,
    const float* __restrict__ e1W, const float* __restrict__ e1b,
    const float* __restrict__ e1s, const float* __restrict__ e1t,
    const float* __restrict__ c0W, const float* __restrict__ c0b,
    const float* __restrict__ c0s, const float* __restrict__ c0t,
    const float* __restrict__ c1W, const float* __restrict__ c1b,
    const float* __restrict__ c1s, const float* __restrict__ c1t,
    float* __restrict__ xpart)
{
  __shared__ float stage[8][16 * STRIDE_ + 16];
  const int wid = threadIdx.x >> 5;
  const int lane = threadIdx.x & 31;
  const int grp = lane >> 4;
  const int ln  = lane & 15;
  const int gw = blockIdx.x * 8 + wid;          // global wave-tile id
  const int b  = gw / (N_ * JT_);
  const int r0 = gw - b * (N_ * JT_);
  const int i  = r0 / JT_;
  const int jt = r0 - i * JT_;
  const int j  = jt * 16 + ln;

  const float* xi = x + (size_t)(b * N_ + i) * 3;
  const float* xj = x + (size_t)(b * N_ + j) * 3;
  const float xi0 = xi[0], xi1 = xi[1], xi2 = xi[2];
  const float dx0 = xj[0] - xi0, dx1 = xj[1] - xi1, dx2 = xj[2] - xi2;
  const float d2  = dx0 * dx0 + dx1 * dx1 + dx2 * dx2;
  const float* hi = h + (size_t)(b * N_ + i) * D_;
  const float* hj = h + (size_t)(b * N_ + j) * D_;

  float* st   = &stage[wid][0];
  float* wbuf = st + 16 * STRIDE_;

  v8f f2[4];
  edge_e0e1(hj, hi, d2, e0W, e0b, e0s, e0t, e1W, e1b, e1s, e1t, st, grp, ln, f2);

  // ---- GEMM3: m_edge(16x64) @ c0W(64x64) ----
  v8f acc3[4] = {};
#pragma unroll
  for (int s = 0; s < 16; ++s) {
    const int k0 = 4 * s + grp * 2;
    v2f A = *(const v2f*)&st[ln * STRIDE_ + k0];
#pragma unroll
    for (int t = 0; t < 4; ++t) {
      const int c = 16 * t + ln;
      v2f Bf; Bf[0] = c0W[k0 * DM_ + c]; Bf[1] = c0W[(k0 + 1) * DM_ + c];
      acc3[t] = wmma4(A, Bf, acc3[t]);
    }
  }
  // affine+ReLU fused with c1 dot product: per-lane partial row dots
  float pv[8] = {0.f, 0.f, 0.f, 0.f, 0.f, 0.f, 0.f, 0.f};
#pragma unroll
  for (int t = 0; t < 4; ++t) {
    const int c = 16 * t + ln;
    const float bb = c0b[c], sc = c0s[c], sh = c0t[c];
    const float w1 = c1W[c];
#pragma unroll
    for (int v = 0; v < 8; ++v) {
      const float f = fmaxf((acc3[t][v] + bb) * sc + sh, 0.0f);
      pv[v] += f * w1;
    }
  }
  // reduce across the 16 lanes of each half-wave (rows of this group)
#pragma unroll
  for (int m = 1; m < 16; m <<= 1) {
#pragma unroll
    for (int v = 0; v < 8; ++v) pv[v] += __shfl_xor(pv[v], m, 32);
  }
  if (ln == 0) {
#pragma unroll
    for (int v = 0; v < 8; ++v) wbuf[v + 8 * grp] = pv[v];
  }
  lds_fence_wave();

  // lane ln owns edge row ln (x_diff already in regs); count each row once
  const float wpre = wbuf[ln];
  const float wv = fmaxf((wpre + c1b[0]) * c1s[0] + c1t[0], 0.0f);
  const float msk = (grp == 0) ? 1.0f : 0.0f;
  float s0 = msk * wv * dx0, s1 = msk * wv * dx1, s2 = msk * wv * dx2;
#pragma unroll
  for (int m = 1; m < 32; m <<= 1) {
    s0 += __shfl_xor(s0, m, 32);
    s1 += __shfl_xor(s1, m, 32);
    s2 += __shfl_xor(s2, m, 32);
  }
  if (lane == 0) {
    float* o = xpart + (size_t)gw * 3;
    o[0] = s0; o[1] = s1; o[2] = s2;
  }
}

// -------------------------------------------------------------------------
// Kernel 3: per-node pass. Wave per node: recompute e0/e1 on the K=16 gathered
// neighbors, row-sum -> m_node, node layer (96->32), finalize x_update.
// -------------------------------------------------------------------------
__global__ __launch_bounds__(256) void egnn_node_pass(
    const float* __restrict__ x, const float* __restrict__ h,
    const float* __restrict__ e0W, const float* __restrict__ e0b,
    const float* __restrict__ e0s, const float* __restrict__ e0t,
    const float* __restrict__ e1W, const float* __restrict__ e1b,
    const float* __restrict__ e1s, const float* __restrict__ e1t,
    const float* __restrict__ n0W, const float* __restrict__ n0b,
    const float* __restrict__ n0s, const float* __restrict__ n0t,
    const int* __restrict__ idx, const float* __restrict__ xpart,
    float* __restrict__ out_x, float* __restrict__ out_h)
{
  __shared__ float stage[8][16 * STRIDE_ + 16];
  __shared__ float mbuf[8][DM_];
  const int wid = threadIdx.x >> 5;
  const int lane = threadIdx.x & 31;
  const int grp = lane >> 4;
  const int ln  = lane & 15;
  const int g = blockIdx.x * 8 + wid;           // node id
  const int b = g / N_, i = g - b * N_;

  const int j = idx[(size_t)(b * N_ + i) * K_ + ln];   // row ln = neighbor ln
  const float* xi = x + (size_t)(b * N_ + i) * 3;
  const float* xj = x + (size_t)(b * N_ + j) * 3;
  const float dx0 = xj[0] - xi[0], dx1 = xj[1] - xi[1], dx2 = xj[2] - xi[2];
  const float d2  = dx0 * dx0 + dx1 * dx1 + dx2 * dx2;
  const float* hi = h + (size_t)(b * N_ + i) * D_;
  const float* hj = h + (size_t)(b * N_ + j) * D_;

  v8f f2[4];
  edge_e0e1(hj, hi, d2, e0W, e0b, e0s, e0t, e1W, e1b, e1s, e1t,
            &stage[wid][0], grp, ln, f2);

  // m_node[c] = sum over the 16 gathered rows of m_edge[:, c]
#pragma unroll
  for (int t = 0; t < 4; ++t) {
    float sum = 0.0f;
#pragma unroll
    for (int v = 0; v < 8; ++v) sum += f2[t][v];      // this group's 8 rows
    sum += __shfl_xor(sum, 16, 32);                    // + other group's 8 rows
    if (grp == 0) mbuf[wid][16 * t + ln] = sum;
  }
  lds_fence_wave();

  // node layer: lane = output channel c in [0,32)
  const int c = lane;
  float acc = 0.0f;
#pragma unroll
  for (int k = 0; k < D_; ++k)  acc += hi[k] * n0W[k * D_ + c];
#pragma unroll
  for (int k = 0; k < DM_; ++k) acc += mbuf[wid][k] * n0W[(D_ + k) * D_ + c];
  const float hu = fmaxf((acc + n0b[c]) * n0s[c] + n0t[c], 0.0f);
  out_h[(size_t)(b * N_ + i) * D_ + c] = hu;

  // x_update = x + sum of the 48 deterministic tile partials
  if (lane < 3) {
    float sacc = xi[lane];
    for (int t = 0; t < JT_; ++t)
      sacc += xpart[((size_t)(b * N_ + i) * JT_ + t) * 3 + lane];
    out_x[(size_t)(b * N_ + i) * 3 + lane] = sacc;
  }
}

// -------------------------------------------------------------------------
extern "C" void kernel_launch(void* const* d_in, const int* in_sizes, int n_in,
                              void* d_out, int out_size, void* d_ws, size_t ws_size,
                              hipStream_t stream)
{
  const float* x   = (const float*)d_in[0];
  const float* h   = (const float*)d_in[1];
  const float* e0W = (const float*)d_in[2];
  const float* e0b = (const float*)d_in[3];
  const float* e0s = (const float*)d_in[4];
  const float* e0t = (const float*)d_in[5];
  const float* e1W = (const float*)d_in[6];
  const float* e1b = (const float*)d_in[7];
  const float* e1s = (const float*)d_in[8];
  const float* e1t = (const float*)d_in[9];
  const float* c0W = (const float*)d_in[10];
  const float* c0b = (const float*)d_in[11];
  const float* c0s = (const float*)d_in[12];
  const float* c0t = (const float*)d_in[13];
  const float* c1W = (const float*)d_in[14];
  const float* c1b = (const float*)d_in[15];
  const float* c1s = (const float*)d_in[16];
  const float* c1t = (const float*)d_in[17];
  const float* n0W = (const float*)d_in[18];
  const float* n0b = (const float*)d_in[19];
  const float* n0s = (const float*)d_in[20];
  const float* n0t = (const float*)d_in[21];

  // workspace layout: [ idx (B*N*K int) | xpart (B*N*JT*3 f32) ]
  int* idx = (int*)d_ws;
  size_t off = ((size_t)B_ * N_ * K_ * sizeof(int) + 255) & ~(size_t)255;
  float* xpart = (float*)((char*)d_ws + off);

  float* out_x = (float*)d_out;                       // (B,N,3)
  float* out_h = out_x + (size_t)B_ * N_ * 3;         // (B,N,32)

  egnn_knn<<<B_ * N_ / 4, 128, 0, stream>>>(x, idx);

  egnn_edge_pass<<<B_ * N_ * JT_ / 8, 256, 0, stream>>>(
      x, h, e0W, e0b, e0s, e0t, e1W, e1b, e1s, e1t,
      c0W, c0b, c0s, c0t, c1W, c1b, c1s, c1t, xpart);

  egnn_node_pass<<<B_ * N_ / 8, 256, 0, stream>>>(
      x, h, e0W, e0b, e0s, e0t, e1W, e1b, e1s, e1t,
      n0W, n0b, n0s, n0t, idx, xpart, out_x, out_h);
}